// Attention_13494787244748
// MI455X (gfx1250) — compile-verified
//
#include <hip/hip_runtime.h>
#include <math.h>

// ---------------------------------------------------------------------------
// XCA channel attention for MI455X (gfx1250, wave32).
// All GEMM-shaped work uses V_WMMA_F32_16X16X4_F32 (the native fp32 WMMA).
//
// Workspace layout (floats):
//   qkv_pw : B*576*N   (pointwise conv output)
//   qkv_dw : B*576*N   (after depthwise 3x3)  -> q | k | v channel blocks
//   norms  : B*384     (||q_c||, ||k_d|| per batch)
//   attn   : B*6*32*32 (post-softmax)
//   Mf     : B*192*192 (proj_w folded with attn)
// ---------------------------------------------------------------------------

typedef float v2f __attribute__((ext_vector_type(2)));
typedef float f4  __attribute__((ext_vector_type(4)));
typedef float v8f __attribute__((ext_vector_type(8)));

#define WMMA_F32X4(a, b, c) \
    __builtin_amdgcn_wmma_f32_16x16x4_f32(false, (a), false, (b), (short)0, (c), false, false)

// ---------------------------------------------------------------------------
// fp32 WMMA GEMM with LDS-staged, double-buffered B operand.
//   C[b] = A[b] (MxK, row major, lda) * B[b] (KxN, row stride N)
// Block = 128 threads (4 waves) covering 64(M) x 64(N); each wave owns a
// 16x64 strip with 4 accumulators.  K consumed in chunks of 16:
//   - B chunk (16K x 64N) loaded coalesced as 2 float4/thread, stored
//     transposed into LDS [n][k] with stride 18 -> every fragment is an
//     8B-aligned ds_load_b64 at (base + 16-bit immediate), conflict-free reads
//   - all 16 B fragments + 4 A fragments of a chunk preloaded into distinct
//     registers before the 16-WMMA burst (one wait per burst, not per WMMA)
// grid = (N/64, M/64, batches);  requires K % 16 == 0, M % 64 == 0, N % 64 == 0
// ---------------------------------------------------------------------------
__global__ __launch_bounds__(128) void gemm_f32_wmma(
    const float* __restrict__ A, int lda, long sA,
    const float* __restrict__ Bm, long sB,
    float* __restrict__ Cm, long sC,
    int K, int N)
{
    const int tid  = threadIdx.x;
    const int lane = tid & 31;
    const int wave = tid >> 5;
    const int bat  = blockIdx.z;
    A  += (size_t)bat * sA;
    Bm += (size_t)bat * sB;
    Cm += (size_t)bat * sC;

    const int m0   = blockIdx.y * 64 + wave * 16;
    const int n0   = blockIdx.x * 64;
    const int row  = lane & 15;                 // M index (A,C) / N index (B)
    const int koff = (lane < 16) ? 0 : 2;       // K sub-offset per lane half

    // double-buffered transposed B chunk: Bt[buf][n(64)][k(16)], stride 18
    // (even stride keeps every [row][4u+koff] fragment 8B-aligned for b64)
    __shared__ float Bt[2][64 * 18];

    // cooperative-load mapping: float4 #j of thread t covers
    //   lin = (j*128 + t)*4 ; k = lin>>6 ; n = lin&63   (kB1 = kB0+8, same n)
    const int kB0 = (tid * 4) >> 6;
    const int nB0 = (tid * 4) & 63;

    const float* gp0 = Bm + (size_t)kB0 * N + n0 + nB0;
    const float* gp1 = gp0 + (size_t)8 * N;

    v8f acc[4] = {{}, {}, {}, {}};
    const float* arow = A + (size_t)(m0 + row) * lda + koff;

    const int nchunks = K >> 4;

    // ---- preload + stage chunk 0 ----
    f4 r0 = *(const f4*)gp0;
    f4 r1 = *(const f4*)gp1;
    gp0 += (size_t)16 * N;
    gp1 += (size_t)16 * N;
#pragma unroll
    for (int i = 0; i < 4; ++i) {
        Bt[0][(nB0 + i) * 18 + kB0]     = r0[i];
        Bt[0][(nB0 + i) * 18 + kB0 + 8] = r1[i];
    }
    __syncthreads();

    for (int c = 0; c < nchunks; ++c) {
        const bool more = (c + 1 < nchunks);
        f4 p0, p1;
        if (more) {
            p0 = *(const f4*)gp0;
            p1 = *(const f4*)gp1;
            // keep L2 ahead of the stream (speculative, dropped if invalid)
            __builtin_prefetch(gp0 + (size_t)16 * N, 0, 0);
            __builtin_prefetch(gp1 + (size_t)16 * N, 0, 0);
            gp0 += (size_t)16 * N;
            gp1 += (size_t)16 * N;
        }

        // ---- compute chunk c out of Bt[c&1] ----
        // per-lane base: all 16 fragments are base + compile-time offset
        const float* btrow = &Bt[c & 1][0] + row * 18 + koff;
        const float* ac    = arow + (c << 4);

        v2f afrag[4];
        v2f bfrag[4][4];
#pragma unroll
        for (int u = 0; u < 4; ++u)
            afrag[u] = *(const v2f*)(ac + (u << 2));
#pragma unroll
        for (int u = 0; u < 4; ++u)
#pragma unroll
            for (int j = 0; j < 4; ++j)
                bfrag[u][j] = *(const v2f*)(btrow + j * (16 * 18) + (u << 2));

#pragma unroll
        for (int u = 0; u < 4; ++u)
#pragma unroll
            for (int j = 0; j < 4; ++j)
                acc[j] = WMMA_F32X4(afrag[u], bfrag[u][j], acc[j]);

        if (more) {
            const int nb = (c + 1) & 1;
#pragma unroll
            for (int i = 0; i < 4; ++i) {
                Bt[nb][(nB0 + i) * 18 + kB0]     = p0[i];
                Bt[nb][(nB0 + i) * 18 + kB0 + 8] = p1[i];
            }
        }
        __syncthreads();
    }

    // C/D layout: VGPR i -> M = m0 + i + (lane<16 ? 0 : 8), N = n0 + (lane&15)
    const int mb = m0 + ((lane < 16) ? 0 : 8);
#pragma unroll
    for (int i = 0; i < 8; ++i) {
        float* crow = Cm + (size_t)(mb + i) * N + n0 + row;
        crow[0]  = acc[0][i];
        crow[16] = acc[1][i];
        crow[32] = acc[2][i];
        crow[48] = acc[3][i];
    }
}

// ---------------------------------------------------------------------------
// Depthwise 3x3, padding SAME, groups == channels (576).
// ---------------------------------------------------------------------------
__global__ __launch_bounds__(256) void dwconv3x3_kernel(
    const float* __restrict__ in, const float* __restrict__ w,
    float* __restrict__ out, int H, int W, long total)
{
    long idx = (long)blockIdx.x * 256 + threadIdx.x;
    if (idx >= total) return;
    const int  x  = (int)(idx % W);
    const int  y  = (int)((idx / W) % H);
    const long ch = idx / ((long)W * H);          // flat (b*576 + c)
    const int  c  = (int)(ch % 576);

    const float* wp   = w + c * 9;
    const float* base = in + ch * (long)H * W;

    float s = 0.f;
#pragma unroll
    for (int dy = -1; dy <= 1; ++dy) {
        const int yy = y + dy;
        if ((unsigned)yy >= (unsigned)H) continue;
#pragma unroll
        for (int dx = -1; dx <= 1; ++dx) {
            const int xx = x + dx;
            if ((unsigned)xx >= (unsigned)W) continue;
            s += wp[(dy + 1) * 3 + (dx + 1)] * base[(long)yy * W + xx];
        }
    }
    out[idx] = s;
}

// ---------------------------------------------------------------------------
// Row L2 norms for q (channels 0..191) and k (192..383) of qkv_dw.
// One 256-thread block per (b, row).  grid = B*384.
// ---------------------------------------------------------------------------
__global__ __launch_bounds__(256) void sumsq_kernel(
    const float* __restrict__ qkv, float* __restrict__ norms, int N)
{
    const int b = blockIdx.x / 384;
    const int r = blockIdx.x % 384;
    const float* p = qkv + ((size_t)b * 576 + r) * N;

    float s = 0.f;
    for (int i = threadIdx.x * 4; i < N; i += 256 * 4) {
        const f4 v = *(const f4*)(p + i);
        s += v[0] * v[0] + v[1] * v[1] + v[2] * v[2] + v[3] * v[3];
    }
    __shared__ float red[256];
    red[threadIdx.x] = s;
    __syncthreads();
    for (int off = 128; off > 0; off >>= 1) {
        if (threadIdx.x < off) red[threadIdx.x] += red[threadIdx.x + off];
        __syncthreads();
    }
    if (threadIdx.x == 0) norms[blockIdx.x] = sqrtf(red[0]);
}

// ---------------------------------------------------------------------------
// Per-(b,h): S = q(32xN) * k(32xN)^T via WMMA with K split across 8 waves,
// LDS reduce, then scale by 1/(||q||*||k||)*temperature and softmax over d.
// K consumed 16 at a time: 16 b64 loads batched ahead of 16 WMMAs so waits
// amortize instead of one full wait per WMMA.
// grid = B*heads, block = 256 threads (8 waves).
// ---------------------------------------------------------------------------
__global__ __launch_bounds__(256) void attn_softmax_kernel(
    const float* __restrict__ qkv, const float* __restrict__ norms,
    const float* __restrict__ temp, float* __restrict__ attn, int N)
{
    const int bh = blockIdx.x;
    const int b  = bh / 6;
    const int h  = bh % 6;
    const float* q  = qkv + ((size_t)b * 576 +       h * 32) * N;
    const float* km = qkv + ((size_t)b * 576 + 192 + h * 32) * N;

    const int lane = threadIdx.x & 31;
    const int wave = threadIdx.x >> 5;
    const int row  = lane & 15;
    const int koff = (lane < 16) ? 0 : 2;

    v8f acc0 = {}, acc1 = {}, acc2 = {}, acc3 = {};

    const int chunk = N / 8;
    const int kbeg  = wave * chunk;
    const int kend  = kbeg + chunk;

    const float* q0 = q  + (size_t)row * N        + koff;
    const float* q1 = q  + (size_t)(row + 16) * N + koff;
    const float* k0 = km + (size_t)row * N        + koff;
    const float* k1 = km + (size_t)(row + 16) * N + koff;

    for (int kk = kbeg; kk < kend; kk += 16) {
        v2f a0[4], a1[4], b0[4], b1[4];
#pragma unroll
        for (int u = 0; u < 4; ++u) {
            const int o = kk + (u << 2);
            a0[u] = *(const v2f*)(q0 + o);
            a1[u] = *(const v2f*)(q1 + o);
            b0[u] = *(const v2f*)(k0 + o);
            b1[u] = *(const v2f*)(k1 + o);
        }
#pragma unroll
        for (int u = 0; u < 4; ++u) {
            acc0 = WMMA_F32X4(a0[u], b0[u], acc0);   // c 0..15 , d 0..15
            acc1 = WMMA_F32X4(a0[u], b1[u], acc1);   // c 0..15 , d 16..31
            acc2 = WMMA_F32X4(a1[u], b0[u], acc2);   // c 16..31, d 0..15
            acc3 = WMMA_F32X4(a1[u], b1[u], acc3);   // c 16..31, d 16..31
        }
    }

    __shared__ float S[32][33];
    for (int i = threadIdx.x; i < 32 * 33; i += 256) (&S[0][0])[i] = 0.f;
    __syncthreads();

    const int cb = (lane < 16) ? 0 : 8;
#pragma unroll
    for (int i = 0; i < 8; ++i) {
        atomicAdd(&S[cb + i][row],           acc0[i]);
        atomicAdd(&S[cb + i][16 + row],      acc1[i]);
        atomicAdd(&S[16 + cb + i][row],      acc2[i]);
        atomicAdd(&S[16 + cb + i][16 + row], acc3[i]);
    }
    __syncthreads();

    if (threadIdx.x < 32) {
        const int   c  = threadIdx.x;
        const float qn = fmaxf(norms[b * 384 + h * 32 + c], 1e-12f);
        const float t  = temp[h];

        float vals[32];
        float mx = -INFINITY;
#pragma unroll
        for (int d = 0; d < 32; ++d) {
            const float kn = fmaxf(norms[b * 384 + 192 + h * 32 + d], 1e-12f);
            const float v  = S[c][d] / (qn * kn) * t;
            vals[d] = v;
            mx = fmaxf(mx, v);
        }
        float ssum = 0.f;
#pragma unroll
        for (int d = 0; d < 32; ++d) {
            const float e = __expf(vals[d] - mx);
            vals[d] = e;
            ssum += e;
        }
        const float inv = 1.f / ssum;
        float* ap = attn + ((size_t)bh * 32 + c) * 32;
#pragma unroll
        for (int d = 0; d < 32; ++d) ap[d] = vals[d] * inv;
    }
}

// ---------------------------------------------------------------------------
// Fold projection into attention:  Mf[b][o][h*32+d] =
//   sum_c proj_w[o][h*32+c] * attn[b][h][c][d]       (tiny: 8*192*192*32 MACs)
// ---------------------------------------------------------------------------
__global__ __launch_bounds__(256) void fuse_proj_kernel(
    const float* __restrict__ proj_w, const float* __restrict__ attn,
    float* __restrict__ Mf)
{
    const int idx = blockIdx.x * 256 + threadIdx.x;
    if (idx >= 8 * 192 * 192) return;
    const int d192 = idx % 192;
    const int o    = (idx / 192) % 192;
    const int b    = idx / (192 * 192);
    const int h    = d192 / 32;
    const int dd   = d192 % 32;

    const float* ap = attn + (((size_t)b * 6 + h) * 32) * 32 + dd;  // stride 32 per c
    const float* pw = proj_w + (size_t)o * 192 + h * 32;
    float s = 0.f;
#pragma unroll
    for (int c = 0; c < 32; ++c) s += pw[c] * ap[c * 32];
    Mf[idx] = s;
}

// ---------------------------------------------------------------------------
extern "C" void kernel_launch(void* const* d_in, const int* in_sizes, int n_in,
                              void* d_out, int out_size, void* d_ws, size_t ws_size,
                              hipStream_t stream)
{
    (void)in_sizes; (void)n_in; (void)out_size; (void)ws_size;

    const float* x        = (const float*)d_in[0];  // [8,192,128,128]
    const float* qkv_w    = (const float*)d_in[1];  // [576,192]
    const float* qkv_dw_w = (const float*)d_in[2];  // [576,1,3,3]
    const float* proj_w   = (const float*)d_in[3];  // [192,192]
    const float* temp     = (const float*)d_in[4];  // [6]
    float*       out      = (float*)d_out;          // [8,192,128,128]

    const int B = 8, H = 128, W = 128, N = H * W;   // N = 16384

    float* ws      = (float*)d_ws;
    float* qkv_pw  = ws;                                  // B*576*N
    float* qkv_dw  = qkv_pw + (size_t)B * 576 * N;        // B*576*N
    float* norms   = qkv_dw + (size_t)B * 576 * N;        // B*384
    float* attn    = norms  + (size_t)B * 384;            // B*6*32*32
    float* Mf      = attn   + (size_t)B * 6 * 32 * 32;    // B*192*192

    // 1) qkv = qkv_w (576x192) x x[b] (192xN)
    {
        dim3 grid(N / 64, 576 / 64, B);
        gemm_f32_wmma<<<grid, 128, 0, stream>>>(
            qkv_w, 192, 0L,
            x, (long)192 * N,
            qkv_pw, (long)576 * N,
            192, N);
    }

    // 2) depthwise 3x3 SAME
    {
        const long total = (long)B * 576 * N;
        dwconv3x3_kernel<<<(unsigned)((total + 255) / 256), 256, 0, stream>>>(
            qkv_pw, qkv_dw_w, qkv_dw, H, W, total);
    }

    // 3) q/k row norms
    sumsq_kernel<<<B * 384, 256, 0, stream>>>(qkv_dw, norms, N);

    // 4) Gram + scale + softmax
    attn_softmax_kernel<<<B * 6, 256, 0, stream>>>(qkv_dw, norms, temp, attn, N);

    // 5) fold proj_w with attn
    fuse_proj_kernel<<<(8 * 192 * 192 + 255) / 256, 256, 0, stream>>>(
        proj_w, attn, Mf);

    // 6) out[b] = Mf[b] (192x192) x v[b] (192xN);  v = qkv_dw channels 384..575
    {
        dim3 grid(N / 64, 192 / 64, B);
        gemm_f32_wmma<<<grid, 128, 0, stream>>>(
            Mf, 192, (long)192 * 192,
            qkv_dw + (size_t)384 * N, (long)576 * N,
            out, (long)192 * N,
            192, N);
    }
}